// SpectralConv1d_41549513622020
// MI455X (gfx1250) — compile-verified
//
#include <hip/hip_runtime.h>
#include <math.h>

#if defined(__HIP_DEVICE_COMPILE__) && !__has_builtin(__builtin_amdgcn_wmma_f32_16x16x4_f32)
#error "gfx1250 f32 WMMA builtin __builtin_amdgcn_wmma_f32_16x16x4_f32 not available (device pass)"
#endif

#if defined(__HIP_DEVICE_COMPILE__) && \
    __has_builtin(__builtin_amdgcn_global_load_async_to_lds_b128) && \
    __has_builtin(__builtin_amdgcn_s_wait_asynccnt)
#define USE_ASYNC_LDS 1
#else
#define USE_ASYNC_LDS 0
#endif

typedef float v2f __attribute__((ext_vector_type(2)));
typedef float v8f __attribute__((ext_vector_type(8)));
typedef int   v4i __attribute__((ext_vector_type(4)));
typedef __attribute__((address_space(1))) v4i* gas_v4i_p;   // global int4*
typedef __attribute__((address_space(3))) v4i* las_v4i_p;   // LDS int4*

#define B_    32
#define CIN   64
#define COUT  64
#define NN    4096
#define LOGN  12
#define MODES 1024
#define KOUT  2049          // N/2 + 1
#define LDB2  2112          // 33*64, padded ld for cas2 (zeros past 2049)
#define NGRP  (LDB2 / 64)   // 33 column groups of 64
#define LDSLD 72            // LDS row stride (floats): 64 + 8 pad, bank-safe
#define PI_F  3.14159265358979323846f

// ---------------------------------------------------------------------------
// Stage 1: recursive_fht, implemented exactly as the reference recursion:
// iterative radix-2 DIT with twiddle t[j] = cos(pi*j/L) + sin(pi*j/L).
// One (b,c) row of 4096 floats per workgroup, in LDS. Keep first 1024 modes.
// ---------------------------------------------------------------------------
__global__ void fht_rows_kernel(const float* __restrict__ x,
                                float* __restrict__ xht) {
  __shared__ float s[NN];
  const int row = blockIdx.x;                       // row = b*CIN + c
  const float* xr = x + (size_t)row * NN;

  for (int i = threadIdx.x; i < NN; i += blockDim.x) {
    unsigned r = __brev((unsigned)i) >> (32 - LOGN);
    s[i] = xr[r];
  }
  __syncthreads();

  for (int lg = 0; lg < LOGN; ++lg) {
    const int L = 1 << lg;
    for (int u = threadIdx.x; u < NN / 2; u += blockDim.x) {
      const int j   = u & (L - 1);
      const int blk = u >> lg;
      const int i0  = (blk << (lg + 1)) + j;
      const float ang = PI_F * (float)j / (float)L;   // 2*pi*j / (2L)
      const float t   = cosf(ang) + sinf(ang);
      const float a  = s[i0];
      const float bb = s[i0 + L];
      s[i0]     = a + t * bb;
      s[i0 + L] = a - t * bb;
    }
    __syncthreads();
  }

  for (int k = threadIdx.x; k < MODES; k += blockDim.x)
    xht[(size_t)row * MODES + k] = s[k];
}

// ---------------------------------------------------------------------------
// cas matrix for idht, padded to LDB2 columns (zeros >= KOUT)
// ---------------------------------------------------------------------------
__global__ void cas2_init_kernel(float* __restrict__ cas2) {
  const int idx = blockIdx.x * blockDim.x + threadIdx.x;
  if (idx >= MODES * LDB2) return;
  const int k = idx / LDB2;
  const int n = idx % LDB2;
  float v = 0.0f;
  if (n < KOUT) {
    const float ang = 6.2831853071795865f * (float)(k * n) / (float)KOUT;
    v = cosf(ang) + sinf(ang);
  }
  cas2[idx] = v;
}

// ---------------------------------------------------------------------------
// Stage 2: per-mode mixing (compl_mul1d simplification):
//   Y[b,o,k] = 0.5*( sum_i X[b,i,k]*(W[i,o,k]+W[63-i,o,k])
//                  + sum_i X[31-b,i,k]*(W[i,o,k]-W[63-i,o,k]) )
// One mode per block; 8 waves cover the 32x64 (b,o) output as 16x16 WMMA
// tiles, each wave running one f32 WMMA accumulation chain (two WMMAs per step) over i (K=64).
// ---------------------------------------------------------------------------
__global__ void mode_mix_wmma_kernel(const float* __restrict__ xht,
                                     const float* __restrict__ w1,
                                     float* __restrict__ y) {
  const int k    = blockIdx.x;                 // mode
  const int wave = threadIdx.x >> 5;           // 0..7
  const int lane = threadIdx.x & 31;
  const int half = lane >> 4;
  const int l    = lane & 15;
  const int m0   = (wave >> 2) << 4;           // batch tile: 0 or 16
  const int n0   = (wave & 3) << 4;            // cout tile : 0..48

  const int bA  = m0 + l;                      // A row (lanes 0-15 hold M=0..15)
  const int bAf = (B_ - 1) - bA;               // batch-flipped A row

  v8f c = {};
  for (int i0 = 0; i0 < CIN; i0 += 4) {
    const int ia = i0 + 2 * half;
    v2f a1, a2, wp, wm;
    a1.x = xht[((size_t)(bA  * CIN + ia    )) * MODES + k];
    a1.y = xht[((size_t)(bA  * CIN + ia + 1)) * MODES + k];
    a2.x = xht[((size_t)(bAf * CIN + ia    )) * MODES + k];
    a2.y = xht[((size_t)(bAf * CIN + ia + 1)) * MODES + k];

    const int o = n0 + l;
    const float w0  = w1[((size_t)((ia    ) * COUT + o)) * MODES + k];
    const float w0f = w1[((size_t)((63 - ia) * COUT + o)) * MODES + k];
    const float wv1 = w1[((size_t)((ia + 1) * COUT + o)) * MODES + k];
    const float w1f = w1[((size_t)((62 - ia) * COUT + o)) * MODES + k];
    wp.x = w0 + w0f;  wp.y = wv1 + w1f;
    wm.x = w0 - w0f;  wm.y = wv1 - w1f;

    c = __builtin_amdgcn_wmma_f32_16x16x4_f32(false, a1, false, wp,
                                              (short)0, c, false, false);
    c = __builtin_amdgcn_wmma_f32_16x16x4_f32(false, a2, false, wm,
                                              (short)0, c, false, false);
  }

#pragma unroll
  for (int r = 0; r < 8; ++r) {
    const int b = m0 + r + 8 * half;
    const int o = n0 + l;
    y[((size_t)(b * COUT + o)) * MODES + k] = 0.5f * c[r];
  }
}

// ---------------------------------------------------------------------------
// Stage 3: idht GEMM  out = (Y @ cas2) / KOUT
//   Y: 2048 x 1024 ; cas2: 1024 x LDB2 ; out: 2048 x 2049
// Block = 8 waves sharing one 64-column group g; each wave owns one 16-row
// tile and 4 WMMA accumulation chains (16x64 strip). The cas2 chunk
// (32 k-rows x 64 cols) is staged in LDS once per block — via CDNA5 async
// load-to-LDS (ASYNCcnt) when the builtin is available.
// ---------------------------------------------------------------------------
__global__ void idht_gemm_wmma_kernel(const float* __restrict__ A,
                                      const float* __restrict__ Bm,
                                      float* __restrict__ C) {
  __shared__ float bs[32 * LDSLD];

  const int g    = blockIdx.x / 16;            // column group 0..32
  const int rm   = blockIdx.x % 16;            // row-tile octet
  const int wave = threadIdx.x >> 5;           // 0..7
  const int lane = threadIdx.x & 31;
  const int half = lane >> 4;
  const int l    = lane & 15;
  const int m0   = (rm * 8 + wave) * 16;       // output row tile base (0..2032)
  const int gbase = g * 64;                    // output col group base

  const float* arow = A + (size_t)(m0 + l) * MODES;

  v8f acc[4];
#pragma unroll
  for (int q = 0; q < 4; ++q) acc[q] = (v8f){};

  for (int k0 = 0; k0 < MODES; k0 += 32) {
    // ---- stage cas2[k0..k0+32) x [gbase..gbase+64) into LDS ----
#pragma unroll
    for (int t = 0; t < 2; ++t) {
      const int idx = (int)threadIdx.x + t * 256;  // 0..511 float4 slots
      const int r   = idx >> 4;                    // k row in chunk 0..31
      const int c4  = (idx & 15) << 2;             // col*4 within group
      const float* gp = Bm + (size_t)(k0 + r) * LDB2 + gbase + c4;
      float* lp = &bs[r * LDSLD + c4];
#if USE_ASYNC_LDS
      __builtin_amdgcn_global_load_async_to_lds_b128(
          (gas_v4i_p)gp, (las_v4i_p)lp, 0, 0);
#else
      *(float4*)lp = *(const float4*)gp;
#endif
    }
#if USE_ASYNC_LDS
    __builtin_amdgcn_s_wait_asynccnt(0);
#endif
    __syncthreads();

    // ---- 8 k-steps of 4, each feeding 4 WMMA chains ----
#pragma unroll
    for (int kk = 0; kk < 32; kk += 4) {
      const v2f a = *(const v2f*)(arow + k0 + kk + 2 * half);
#pragma unroll
      for (int q = 0; q < 4; ++q) {
        v2f b;
        b.x = bs[(kk + 2 * half    ) * LDSLD + q * 16 + l];
        b.y = bs[(kk + 2 * half + 1) * LDSLD + q * 16 + l];
        acc[q] = __builtin_amdgcn_wmma_f32_16x16x4_f32(false, a, false, b,
                                                       (short)0, acc[q],
                                                       false, false);
      }
    }
    __syncthreads();
  }

  const float inv = 1.0f / (float)KOUT;
#pragma unroll
  for (int q = 0; q < 4; ++q) {
#pragma unroll
    for (int r = 0; r < 8; ++r) {
      const int row = m0 + r + 8 * half;
      const int col = gbase + q * 16 + l;
      if (col < KOUT)
        C[(size_t)row * KOUT + col] = acc[q][r] * inv;
    }
  }
}

// ---------------------------------------------------------------------------
extern "C" void kernel_launch(void* const* d_in, const int* in_sizes, int n_in,
                              void* d_out, int out_size, void* d_ws, size_t ws_size,
                              hipStream_t stream) {
  const float* x  = (const float*)d_in[0];   // (32, 64, 4096)
  const float* w1 = (const float*)d_in[1];   // (64, 64, 1024)
  // d_in[2] = modes1 (compile-time constant 1024 here)
  float* out = (float*)d_out;                // (32, 64, 2049)

  float* ws   = (float*)d_ws;
  float* xht  = ws;                                    // 2048*1024 f32 (8 MB)
  float* y    = ws + (size_t)(B_ * CIN) * MODES;       // 2048*1024 f32 (8 MB)
  float* cas2 = y  + (size_t)(B_ * COUT) * MODES;      // 1024*2112 f32 (~8.7 MB)

  // 1) exact recursive-FHT butterflies, keep first 1024 modes
  fht_rows_kernel<<<B_ * CIN, 256, 0, stream>>>(x, xht);

  // 2) idht cas matrix
  cas2_init_kernel<<<(MODES * LDB2 + 255) / 256, 256, 0, stream>>>(cas2);

  // 3) per-mode mixing, WMMA f32
  mode_mix_wmma_kernel<<<MODES, 256, 0, stream>>>(xht, w1, y);

  // 4) idht GEMM, WMMA f32: 128 row tiles x 33 column groups, 8 waves/block
  idht_gemm_wmma_kernel<<<16 * NGRP, 256, 0, stream>>>(y, cas2, out);
}